// DecoderIter_41772851920903
// MI455X (gfx1250) — compile-verified
//
#include <hip/hip_runtime.h>
#include <hip/hip_bf16.h>
#include <math.h>

// ---------------------------------------------------------------------------
// Tacotron2 decoder step on MI455X (gfx1250, wave32).
// Memory-bound (~410MB stream, ~18us roofline @23.3TB/s).
//  - GEMMs: V_WMMA_F32_16X16X4_F32, 2x2 tiles/wave (4 WMMA per 4 frag loads).
//  - ctx matvec (268MB 'memory' stream): TDM tensor_load_to_lds double-buffer.
// ---------------------------------------------------------------------------

typedef __attribute__((ext_vector_type(2))) float v2f;
typedef __attribute__((ext_vector_type(8))) float v8f;
typedef unsigned int v4ui __attribute__((ext_vector_type(4)));
typedef int v8i __attribute__((ext_vector_type(8)));
typedef int v4i __attribute__((ext_vector_type(4)));

#define BATCH 128
#define TLEN  1024
#define NMEL  80
#define PRE   256
#define ENC   512
#define RNN   1024
#define ATTD  128

// ---------------------------------------------------------------------------
// f32 WMMA GEMM, 2x2 register tiling: each wave computes a 32x32 C tile.
// C[M,N] (+)= A[M,K] * W[N,K]^T (+bias)(+relu). M,N % 32 == 0, K % 4 == 0.
// ---------------------------------------------------------------------------
__global__ void wmma_gemm_2x2(const float* __restrict__ A, int lda,
                              const float* __restrict__ W, int ldw,
                              float* __restrict__ C, int ldc,
                              int M, int N, int K,
                              const float* __restrict__ bias,
                              int accumulate, int relu) {
    int tilesN = N >> 5;
    int tiles  = (M >> 5) * tilesN;
    int tile   = blockIdx.x * 8 + (threadIdx.x >> 5);
    if (tile >= tiles) return;                    // uniform whole-wave exit
    int m0   = (tile / tilesN) << 5;
    int n0   = (tile % tilesN) << 5;
    int lane = threadIdx.x & 31;
    int half = lane >> 4, l15 = lane & 15;

    v8f acc[2][2];
    #pragma unroll
    for (int i = 0; i < 2; ++i)
        #pragma unroll
        for (int j = 0; j < 2; ++j) {
            if (accumulate) {
                for (int r = 0; r < 8; ++r)
                    acc[i][j][r] = C[(size_t)(m0 + 16*i + r + 8*half) * ldc
                                     + n0 + 16*j + l15];
            } else if (bias) {
                float bv = bias[n0 + 16*j + l15];
                for (int r = 0; r < 8; ++r) acc[i][j][r] = bv;
            } else {
                for (int r = 0; r < 8; ++r) acc[i][j][r] = 0.f;
            }
        }

    const float* Ap0 = A + (size_t)(m0 + l15) * lda + 2 * half;
    const float* Ap1 = Ap0 + (size_t)16 * lda;
    const float* Wp0 = W + (size_t)(n0 + l15) * ldw + 2 * half;
    const float* Wp1 = Wp0 + (size_t)16 * ldw;
    #pragma unroll 2
    for (int k = 0; k < K; k += 4) {
        v2f a0 = *(const v2f*)(Ap0 + k);
        v2f a1 = *(const v2f*)(Ap1 + k);
        v2f b0 = *(const v2f*)(Wp0 + k);
        v2f b1 = *(const v2f*)(Wp1 + k);
        acc[0][0] = __builtin_amdgcn_wmma_f32_16x16x4_f32(false, a0, false, b0, (short)0, acc[0][0], false, false);
        acc[0][1] = __builtin_amdgcn_wmma_f32_16x16x4_f32(false, a0, false, b1, (short)0, acc[0][1], false, false);
        acc[1][0] = __builtin_amdgcn_wmma_f32_16x16x4_f32(false, a1, false, b0, (short)0, acc[1][0], false, false);
        acc[1][1] = __builtin_amdgcn_wmma_f32_16x16x4_f32(false, a1, false, b1, (short)0, acc[1][1], false, false);
    }

    #pragma unroll
    for (int i = 0; i < 2; ++i)
        #pragma unroll
        for (int j = 0; j < 2; ++j)
            for (int r = 0; r < 8; ++r) {
                float x = acc[i][j][r];
                if (relu) x = fmaxf(x, 0.f);
                C[(size_t)(m0 + 16*i + r + 8*half) * ldc + n0 + 16*j + l15] = x;
            }
}

// 1x1-tile fallback for N % 32 != 0 (the N=80 projection head).
__global__ void wmma_gemm_1x1(const float* __restrict__ A, int lda,
                              const float* __restrict__ W, int ldw,
                              float* __restrict__ C, int ldc,
                              int M, int N, int K,
                              const float* __restrict__ bias,
                              int accumulate, int relu) {
    int tilesN = N >> 4;
    int tiles  = (M >> 4) * tilesN;
    int tile   = blockIdx.x * 8 + (threadIdx.x >> 5);
    if (tile >= tiles) return;
    int m0   = (tile / tilesN) << 4;
    int n0   = (tile % tilesN) << 4;
    int lane = threadIdx.x & 31;
    int half = lane >> 4, l15 = lane & 15;

    v8f acc = {};
    if (accumulate) {
        for (int r = 0; r < 8; ++r)
            acc[r] = C[(size_t)(m0 + r + 8*half) * ldc + n0 + l15];
    } else if (bias) {
        float bv = bias[n0 + l15];
        for (int r = 0; r < 8; ++r) acc[r] = bv;
    }
    const float* Ap = A + (size_t)(m0 + l15) * lda + 2 * half;
    const float* Wp = W + (size_t)(n0 + l15) * ldw + 2 * half;
    #pragma unroll 4
    for (int k = 0; k < K; k += 4) {
        v2f a = *(const v2f*)(Ap + k);
        v2f b = *(const v2f*)(Wp + k);
        acc = __builtin_amdgcn_wmma_f32_16x16x4_f32(false, a, false, b, (short)0, acc, false, false);
    }
    for (int r = 0; r < 8; ++r) {
        float x = acc[r];
        if (relu) x = fmaxf(x, 0.f);
        C[(size_t)(m0 + r + 8*half) * ldc + n0 + l15] = x;
    }
}

// ---------------------------------------------------------------------------
// Prenet dropout mask (deterministic hash Bernoulli(0.5), scale x2).
// ---------------------------------------------------------------------------
__global__ void mask_scale(float* __restrict__ x, int cols, unsigned seed, int n) {
    int idx = blockIdx.x * blockDim.x + threadIdx.x;
    if (idx >= n) return;
    unsigned c = (unsigned)(idx % cols);
    unsigned h = (c * 2654435761u) ^ (seed * 2246822519u);
    h ^= h >> 15; h *= 2654435761u; h ^= h >> 13;
    x[idx] *= (h & 1u) ? 2.0f : 0.0f;
}

__global__ void concat2(float* __restrict__ dst,
                        const float* __restrict__ a, int ac,
                        const float* __restrict__ b, int bc, int n) {
    int idx = blockIdx.x * blockDim.x + threadIdx.x;
    if (idx >= n) return;
    int w = ac + bc;
    int r = idx / w, c = idx - r * w;
    dst[idx] = (c < ac) ? a[(size_t)r * ac + c] : b[(size_t)r * bc + (c - ac)];
}

__global__ void lstm_pointwise(const float* __restrict__ G,
                               const float* __restrict__ bhh,
                               const float* __restrict__ c_prev,
                               float* __restrict__ h_out,
                               float* __restrict__ c_out) {
    int idx = blockIdx.x * blockDim.x + threadIdx.x;
    int b = idx >> 10, j = idx & 1023;
    const float* g = G + (size_t)b * 4096;
    float gi = g[j]        + bhh[j];
    float gf = g[1024 + j] + bhh[1024 + j];
    float gg = g[2048 + j] + bhh[2048 + j];
    float go = g[3072 + j] + bhh[3072 + j];
    float si = 1.f / (1.f + __expf(-gi));
    float sf = 1.f / (1.f + __expf(-gf));
    float so = 1.f / (1.f + __expf(-go));
    float c2 = sf * c_prev[idx] + si * tanhf(gg);
    h_out[idx] = so * tanhf(c2);
    c_out[idx] = c2;
}

// ---------------------------------------------------------------------------
// Fused location attention energies (conv -> LDS -> WMMA proj -> tanh-dot).
// ---------------------------------------------------------------------------
__global__ void attn_energies(const float* __restrict__ aw,
                              const float* __restrict__ awcum,
                              const float* __restrict__ Wconv,  // [32,2,31]
                              const float* __restrict__ Wloc,   // [128,32]
                              const float* __restrict__ q,      // [B,128]
                              const float* __restrict__ pm,     // [B,T,128]
                              const float* __restrict__ v,      // [128]
                              float* __restrict__ energies) {   // [B,T]
    int b  = blockIdx.x >> 6;
    int t0 = (blockIdx.x & 63) << 4;
    __shared__ __align__(16) float locT[16 * 32];
    __shared__ float e16[16];
    int tid = threadIdx.x;
    if (tid < 16) e16[tid] = 0.f;

    for (int i = tid; i < 512; i += 256) {
        int tl = i >> 5, f = i & 31;
        int t = t0 + tl;
        const float* w = Wconv + f * 62;
        float s = 0.f;
        #pragma unroll
        for (int k = 0; k < 31; ++k) {
            int tp = t + k - 15;
            if (tp >= 0 && tp < TLEN) {
                s = fmaf(w[k],      aw[(size_t)b * TLEN + tp], s);
                s = fmaf(w[31 + k], awcum[(size_t)b * TLEN + tp], s);
            }
        }
        locT[tl * 32 + f] = s;
    }
    __syncthreads();

    int wv = tid >> 5, lane = tid & 31, half = lane >> 4, l15 = lane & 15;
    int d0 = wv << 4;
    v8f acc = {};
    #pragma unroll
    for (int k = 0; k < 32; k += 4) {
        v2f a  = *(const v2f*)(&locT[l15 * 32 + k + 2 * half]);
        v2f bb = *(const v2f*)(&Wloc[(d0 + l15) * 32 + k + 2 * half]);
        acc = __builtin_amdgcn_wmma_f32_16x16x4_f32(false, a, false, bb, (short)0, acc, false, false);
    }

    int d = d0 + l15;
    float vd = v[d];
    float qd = q[(size_t)b * ATTD + d];
    #pragma unroll
    for (int r = 0; r < 8; ++r) {
        int t = t0 + r + 8 * half;
        float val = tanhf(acc[r] + qd + pm[((size_t)b * TLEN + t) * ATTD + d]) * vd;
        for (int off = 8; off >= 1; off >>= 1)
            val += __shfl_xor(val, off, 16);
        if (l15 == 0) atomicAdd(&e16[r + 8 * half], val);
    }
    __syncthreads();
    if (tid < 16) energies[(size_t)b * TLEN + t0 + tid] = e16[tid];
}

__global__ void softmax_kernel(const float* __restrict__ energies,
                               const float* __restrict__ awcum_in,
                               float* __restrict__ aw_out,
                               float* __restrict__ awcum_out) {
    __shared__ float red[256];
    int b = blockIdx.x, tid = threadIdx.x;
    const float* e = energies + (size_t)b * TLEN;
    float vbuf[4]; float m = -3.0e38f;
    #pragma unroll
    for (int i = 0; i < 4; ++i) { vbuf[i] = e[tid + (i << 8)]; m = fmaxf(m, vbuf[i]); }
    red[tid] = m; __syncthreads();
    for (int s = 128; s > 0; s >>= 1) {
        if (tid < s) red[tid] = fmaxf(red[tid], red[tid + s]);
        __syncthreads();
    }
    m = red[0]; __syncthreads();
    float sum = 0.f;
    #pragma unroll
    for (int i = 0; i < 4; ++i) { vbuf[i] = __expf(vbuf[i] - m); sum += vbuf[i]; }
    red[tid] = sum; __syncthreads();
    for (int s = 128; s > 0; s >>= 1) {
        if (tid < s) red[tid] += red[tid + s];
        __syncthreads();
    }
    float inv = 1.f / red[0];
    #pragma unroll
    for (int i = 0; i < 4; ++i) {
        int t = tid + (i << 8);
        float a = vbuf[i] * inv;
        aw_out[(size_t)b * TLEN + t] = a;
        awcum_out[(size_t)b * TLEN + t] = awcum_in[(size_t)b * TLEN + t] + a;
    }
}

// ---------------------------------------------------------------------------
// ctx[b,e] = sum_t aw[b,t]*memory[b,t,e]. 268MB stream via Tensor Data Mover:
// 2D tensor (dim0=512 contig, dim1=1024 rows), 16-row/32KB tiles, double-
// buffered in LDS; wave0 issues TDM + s_wait_tensorcnt, DMA overlaps FMA.
// ---------------------------------------------------------------------------
__device__ __forceinline__ void tdm_load_tile(unsigned long long gaddr,
                                              unsigned int lds_addr) {
    v4ui g0;
    g0.x = 1u;                                            // count=1, user mode
    g0.y = lds_addr;                                      // LDS byte address
    g0.z = (unsigned int)(gaddr & 0xFFFFFFFFull);         // global_addr lo
    g0.w = (unsigned int)((gaddr >> 32) & 0x1FFFFFFull)   // global_addr hi
           | 0x80000000u;                                 // type=2 ("image")
    v8i g1;
    g1[0] = 0x20000;                  // wg_mask=0 | data_size=2 (4B)
    g1[1] = (int)(512u << 16);        // tensor_dim0=512 (low16 @ [63:48])
    g1[2] = (int)(1024u << 16);       // dim0_hi=0 | tensor_dim1=1024 low16
    g1[3] = (int)(512u << 16);        // dim1_hi=0 | tile_dim0=512
    g1[4] = 16;                       // tile_dim1=16 | tile_dim2=0
    g1[5] = 512;                      // tensor_dim0_stride lo32
    g1[6] = 0;                        // stride0 hi | stride1 lo
    g1[7] = 0;                        // stride1 hi
    v4i z4 = {};
    v8i z8 = {};
    // clang-23 / therock-10.0 arity: (g0, g1, g2, g3, g4, cpol)
    __builtin_amdgcn_tensor_load_to_lds(g0, g1, z4, z4, z8, 0);
}

__global__ void ctx_kernel_tdm(const float* __restrict__ aw,
                               const float* __restrict__ memory,
                               float* __restrict__ ctx_out) {
    __shared__ __align__(16) float tile[2][16 * ENC];     // 2 x 32KB
    int b = blockIdx.x, tid = threadIdx.x;                // 512 threads
    const float* ab = aw + (size_t)b * TLEN;
    unsigned long long gbase =
        (unsigned long long)(uintptr_t)(memory + (size_t)b * TLEN * ENC);
    bool issuer = (tid < 32);                             // wave 0 drives TDM
    const int NT = TLEN / 16;                             // 64 tiles

    if (issuer) tdm_load_tile(gbase, (unsigned int)(uintptr_t)&tile[0][0]);
    float acc = 0.f;
    for (int it = 0; it < NT; ++it) {
        if (issuer) __builtin_amdgcn_s_wait_tensorcnt(0); // tile 'it' landed
        __syncthreads();                                  // visible to block;
                                                          // prev compute done
        if (issuer && it + 1 < NT)
            tdm_load_tile(gbase + (unsigned long long)(it + 1) * 16 * ENC * 4,
                          (unsigned int)(uintptr_t)&tile[(it + 1) & 1][0]);
        const float* tp = &tile[it & 1][0];
        #pragma unroll
        for (int r = 0; r < 16; ++r)
            acc = fmaf(ab[it * 16 + r], tp[r * ENC + tid], acc);
    }
    ctx_out[(size_t)b * ENC + tid] = acc;
}

__global__ void gate_kernel(const float* __restrict__ hc,
                            const float* __restrict__ Wg,
                            const float* __restrict__ bg,
                            float* __restrict__ gate_out) {
    int b = blockIdx.x * 8 + (threadIdx.x >> 5);
    if (b >= BATCH) return;
    int lane = threadIdx.x & 31;
    const float* h = hc + (size_t)b * (RNN + ENC);
    float acc = 0.f;
    for (int j = lane; j < RNN + ENC; j += 32) acc = fmaf(h[j], Wg[j], acc);
    for (int off = 16; off >= 1; off >>= 1) acc += __shfl_xor(acc, off, 32);
    if (lane == 0) gate_out[b] = acc + bg[0];
}

// ---------------------------------------------------------------------------
static inline void launch_gemm(const float* A, int lda, const float* W, int ldw,
                               float* C, int ldc, int M, int N, int K,
                               const float* bias, int acc, int relu,
                               hipStream_t s) {
    if ((M & 31) == 0 && (N & 31) == 0) {
        int tiles = (M >> 5) * (N >> 5);
        wmma_gemm_2x2<<<(tiles + 7) / 8, 256, 0, s>>>(A, lda, W, ldw, C, ldc,
                                                      M, N, K, bias, acc, relu);
    } else {
        int tiles = (M >> 4) * (N >> 4);
        wmma_gemm_1x1<<<(tiles + 7) / 8, 256, 0, s>>>(A, lda, W, ldw, C, ldc,
                                                      M, N, K, bias, acc, relu);
    }
}

extern "C" void kernel_launch(void* const* d_in, const int* in_sizes, int n_in,
                              void* d_out, int out_size, void* d_ws, size_t ws_size,
                              hipStream_t stream) {
    const float* dec_input = (const float*)d_in[0];
    const float* att_h     = (const float*)d_in[1];
    const float* att_c     = (const float*)d_in[2];
    const float* dec_h     = (const float*)d_in[3];
    const float* dec_c     = (const float*)d_in[4];
    const float* aw_in     = (const float*)d_in[5];
    const float* awcum_in  = (const float*)d_in[6];
    const float* att_ctx   = (const float*)d_in[7];
    const float* memory    = (const float*)d_in[8];
    const float* pm        = (const float*)d_in[9];
    const float* W_pre1    = (const float*)d_in[10];
    const float* W_pre2    = (const float*)d_in[11];
    const float* Wih_att   = (const float*)d_in[12];
    const float* Whh_att   = (const float*)d_in[13];
    const float* bih_att   = (const float*)d_in[14];
    const float* bhh_att   = (const float*)d_in[15];
    const float* W_q       = (const float*)d_in[16];
    const float* W_conv    = (const float*)d_in[17];
    const float* W_loc     = (const float*)d_in[18];
    const float* v_att     = (const float*)d_in[19];
    const float* Wih_dec   = (const float*)d_in[20];
    const float* Whh_dec   = (const float*)d_in[21];
    const float* bih_dec   = (const float*)d_in[22];
    const float* bhh_dec   = (const float*)d_in[23];
    const float* W_proj    = (const float*)d_in[24];
    const float* b_proj    = (const float*)d_in[25];
    const float* W_gate    = (const float*)d_in[26];
    const float* b_gate    = (const float*)d_in[27];

    float* out      = (float*)d_out;
    float* o_decout = out;                                   // [128,80]
    float* o_gate   = o_decout + BATCH * NMEL;               // [128,1]
    float* o_ah     = o_gate + BATCH;                        // [128,1024]
    float* o_ac     = o_ah + BATCH * RNN;
    float* o_dh     = o_ac + BATCH * RNN;
    float* o_dc     = o_dh + BATCH * RNN;
    float* o_aw     = o_dc + BATCH * RNN;                    // [128,1024]
    float* o_awcum  = o_aw + BATCH * TLEN;
    float* o_ctx    = o_awcum + BATCH * TLEN;                // [128,512]

    float* ws = (float*)d_ws;
    size_t off = 0;
    float* x1       = ws + off; off += BATCH * PRE;
    float* x2       = ws + off; off += BATCH * PRE;
    float* cell_in  = ws + off; off += BATCH * (PRE + ENC);
    float* G        = ws + off; off += BATCH * 4 * RNN;
    float* qbuf     = ws + off; off += BATCH * ATTD;
    float* energies = ws + off; off += BATCH * TLEN;
    float* dec_in   = ws + off; off += BATCH * (RNN + ENC);
    float* hc       = ws + off; off += BATCH * (RNN + ENC);

    // ---- prenet ----
    launch_gemm(dec_input, NMEL, W_pre1, NMEL, x1, PRE,
                BATCH, PRE, NMEL, nullptr, 0, 1, stream);
    mask_scale<<<(BATCH * PRE + 255) / 256, 256, 0, stream>>>(x1, PRE, 1u, BATCH * PRE);
    launch_gemm(x1, PRE, W_pre2, PRE, x2, PRE,
                BATCH, PRE, PRE, nullptr, 0, 1, stream);
    mask_scale<<<(BATCH * PRE + 255) / 256, 256, 0, stream>>>(x2, PRE, 2u, BATCH * PRE);

    // ---- attention LSTM ----
    concat2<<<(BATCH * (PRE + ENC) + 255) / 256, 256, 0, stream>>>(
        cell_in, x2, PRE, att_ctx, ENC, BATCH * (PRE + ENC));
    launch_gemm(cell_in, PRE + ENC, Wih_att, PRE + ENC, G, 4 * RNN,
                BATCH, 4 * RNN, PRE + ENC, bih_att, 0, 0, stream);
    launch_gemm(att_h, RNN, Whh_att, RNN, G, 4 * RNN,
                BATCH, 4 * RNN, RNN, nullptr, 1, 0, stream);
    lstm_pointwise<<<(BATCH * RNN + 255) / 256, 256, 0, stream>>>(
        G, bhh_att, att_c, o_ah, o_ac);

    // ---- location attention ----
    launch_gemm(o_ah, RNN, W_q, RNN, qbuf, ATTD,
                BATCH, ATTD, RNN, nullptr, 0, 0, stream);
    attn_energies<<<BATCH * (TLEN / 16), 256, 0, stream>>>(
        aw_in, awcum_in, W_conv, W_loc, qbuf, pm, v_att, energies);
    softmax_kernel<<<BATCH, 256, 0, stream>>>(energies, awcum_in, o_aw, o_awcum);
    ctx_kernel_tdm<<<BATCH, ENC, 0, stream>>>(o_aw, memory, o_ctx);

    // ---- decoder LSTM ----
    concat2<<<(BATCH * (RNN + ENC) + 255) / 256, 256, 0, stream>>>(
        dec_in, o_ah, RNN, o_ctx, ENC, BATCH * (RNN + ENC));
    launch_gemm(dec_in, RNN + ENC, Wih_dec, RNN + ENC, G, 4 * RNN,
                BATCH, 4 * RNN, RNN + ENC, bih_dec, 0, 0, stream);
    launch_gemm(dec_h, RNN, Whh_dec, RNN, G, 4 * RNN,
                BATCH, 4 * RNN, RNN, nullptr, 1, 0, stream);
    lstm_pointwise<<<(BATCH * RNN + 255) / 256, 256, 0, stream>>>(
        G, bhh_dec, dec_c, o_dh, o_dc);

    // ---- heads ----
    concat2<<<(BATCH * (RNN + ENC) + 255) / 256, 256, 0, stream>>>(
        hc, o_dh, RNN, o_ctx, ENC, BATCH * (RNN + ENC));
    launch_gemm(hc, RNN + ENC, W_proj, RNN + ENC, o_decout, NMEL,
                BATCH, NMEL, RNN + ENC, b_proj, 0, 0, stream);
    gate_kernel<<<16, 256, 0, stream>>>(hc, W_gate, b_gate, o_gate);
}